// LDPCNeuralDecoder_71614284693539
// MI455X (gfx1250) — compile-verified
//
#include <hip/hip_runtime.h>
#include <math.h>

#define NUM_ITER   5
#define NODES      8448
#define MAX_NBR    10
#define BATCH      1024

#define AS1 __attribute__((address_space(1)))
#define AS3 __attribute__((address_space(3)))

#if __has_builtin(__builtin_amdgcn_global_load_async_to_lds_b32)
#define HAVE_ASYNC_LDS 1
#else
#define HAVE_ASYNC_LDS 0
#endif

// ---------------- fast math helpers (CDNA5 native where available) ----------
// Fallback branches must parse on BOTH host and device passes.

__device__ __forceinline__ float fast_exp2(float x) {
#if __has_builtin(__builtin_amdgcn_exp2f)
  return __builtin_amdgcn_exp2f(x);
#else
  return exp2f(x);
#endif
}

__device__ __forceinline__ float fast_log2(float x) {
#if __has_builtin(__builtin_amdgcn_logf)
  return __builtin_amdgcn_logf(x);
#else
  return log2f(x);
#endif
}

__device__ __forceinline__ float fast_rcp(float x) {
#if __has_builtin(__builtin_amdgcn_rcpf)
  return __builtin_amdgcn_rcpf(x);
#else
  return 1.0f / x;
#endif
}

// gfx1250 adds V_TANH_F32 on the transcendental unit.
__device__ __forceinline__ float fast_tanh(float x) {
#if __has_builtin(__builtin_amdgcn_tanhf)
  return __builtin_amdgcn_tanhf(x);
#elif __has_builtin(__builtin_amdgcn_tanh_f32)
  return __builtin_amdgcn_tanh_f32(x);
#else
  // tanh(x) = 1 - 2/(exp2(2x*log2e)+1)
  float e = fast_exp2(x * 2.8853900817779268f);
  return 1.0f - 2.0f * fast_rcp(e + 1.0f);
#endif
}

__device__ __forceinline__ void wait_async0() {
#if __has_builtin(__builtin_amdgcn_s_wait_asynccnt)
  __builtin_amdgcn_s_wait_asynccnt(0);
#else
  asm volatile("s_wait_asynccnt 0" ::: "memory");
#endif
}

// tanh(clip(0.5*g, -9.9, 9.9))
__device__ __forceinline__ float th_half(float g) {
  float x = 0.5f * g;
  x = fminf(fmaxf(x, -9.9f), 9.9f);
  return fast_tanh(x);
}

// 2*atanh(clip(p)) = ln2 * log2((1+p)/(1-p))
__device__ __forceinline__ float atanh2(float p) {
  p = fminf(fmaxf(p, -0.999999f), 0.999999f);
  return 0.6931471805599453f * fast_log2((1.0f + p) * fast_rcp(1.0f - p));
}

__device__ __forceinline__ float sigmoidf(float x) {
  return fast_rcp(1.0f + fast_exp2(-1.4426950408889634f * x));
}

// ---------------- kernel A: (B,N) -> (N,B) transpose, fused * w_ch ----------
// Global->LDS stage uses the CDNA5 async-to-LDS pipe (ASYNCcnt) when present:
// no VGPR round trip, no explicit ds_store on the fill side.

__global__ __launch_bounds__(256) void k_transpose_weight(
    const float* __restrict__ in,    // (BATCH, NODES)
    const float* __restrict__ w_ch,  // (NODES)
    float* __restrict__ outT)        // (NODES, BATCH)
{
  __shared__ float tile[32][33];          // pad -> conflict-free ds reads
  const int tn = blockIdx.x * 32;         // node tile base
  const int tb = blockIdx.y * 32;         // batch tile base
  const int lx = threadIdx.x;             // 0..31
  const int ly = threadIdx.y;             // 0..7
#if HAVE_ASYNC_LDS
#pragma unroll
  for (int i = 0; i < 4; ++i) {
    const int b = tb + ly + i * 8;
    __builtin_amdgcn_global_load_async_to_lds_b32(
        (AS1 int*)(in + (size_t)b * NODES + tn + lx),
        (AS3 int*)&tile[ly + i * 8][lx], 0, 0);
  }
  wait_async0();
#else
#pragma unroll
  for (int i = 0; i < 4; ++i) {
    const int b = tb + ly + i * 8;
    tile[ly + i * 8][lx] = in[(size_t)b * NODES + tn + lx];   // coalesced in n
  }
#endif
  __syncthreads();
#pragma unroll
  for (int i = 0; i < 4; ++i) {
    const int n = tn + ly + i * 8;
    outT[(size_t)n * BATCH + tb + lx] = tile[lx][ly + i * 8] * w_ch[n]; // coalesced in b
  }
}

// ---------------- kernel B: one BP iteration in node-major layout -----------
// One block per node; 256 threads x float4 cover BATCH=1024.
// Indices are block-uniform -> scalar loads; every gather is coalesced and
// L2-resident (working set ~138 MB < 192 MB L2).

__global__ __launch_bounds__(256) void k_iter(
    const float* __restrict__ wT,    // weighted_llr (N,B)
    const float* __restrict__ cur,   // var_messages (N,B)
    const float* __restrict__ prev,  // var_messages one iter older (or null)
    float* __restrict__ dst,         // new var_messages (N,B)
    const int* __restrict__ cidx,    // (N, MAX_NBR)
    const float* __restrict__ w_res, // (2, N)
    int has_prev)
{
  const int n = blockIdx.x;
  const int b = threadIdx.x * 4;
  const size_t row = (size_t)n * BATCH + b;

  float p[4] = {1.0f, 1.0f, 1.0f, 1.0f};
#pragma unroll
  for (int k = 0; k < MAX_NBR; ++k) {
    const int idx = cidx[n * MAX_NBR + k];                // uniform -> s_load
    const float4 g = *reinterpret_cast<const float4*>(cur + (size_t)idx * BATCH + b);
    p[0] *= th_half(g.x);
    p[1] *= th_half(g.y);
    p[2] *= th_half(g.z);
    p[3] *= th_half(g.w);
  }

  const float w0 = w_res[n];
  const float w1 = w_res[NODES + n];

  const float4 wv = *reinterpret_cast<const float4*>(wT + row);
  const float4 cv = *reinterpret_cast<const float4*>(cur + row);

  float4 r;
  r.x = wv.x + w0 * cv.x + atanh2(p[0]);
  r.y = wv.y + w0 * cv.y + atanh2(p[1]);
  r.z = wv.z + w0 * cv.z + atanh2(p[2]);
  r.w = wv.w + w0 * cv.w + atanh2(p[3]);

  if (has_prev) {
    const float4 pv = *reinterpret_cast<const float4*>(prev + row);
    r.x += w1 * pv.x;
    r.y += w1 * pv.y;
    r.z += w1 * pv.z;
    r.w += w1 * pv.w;
  }

  *reinterpret_cast<float4*>(dst + row) = r;
}

// ---------------- kernel C: (N,B) -> (B,N) transpose, fused sigmoid ---------

__global__ __launch_bounds__(256) void k_final(
    const float* __restrict__ vT,   // (NODES, BATCH)
    const float* __restrict__ in,   // input_llr (BATCH, NODES)
    float* __restrict__ out)        // soft bits (BATCH, NODES)
{
  __shared__ float tile[32][33];
  const int tn = blockIdx.x * 32;
  const int tb = blockIdx.y * 32;
  const int lx = threadIdx.x;
  const int ly = threadIdx.y;
#if HAVE_ASYNC_LDS
#pragma unroll
  for (int i = 0; i < 4; ++i) {
    const int n = tn + ly + i * 8;
    __builtin_amdgcn_global_load_async_to_lds_b32(
        (AS1 int*)(vT + (size_t)n * BATCH + tb + lx),
        (AS3 int*)&tile[ly + i * 8][lx], 0, 0);
  }
  wait_async0();
#else
#pragma unroll
  for (int i = 0; i < 4; ++i) {
    const int n = tn + ly + i * 8;
    tile[ly + i * 8][lx] = vT[(size_t)n * BATCH + tb + lx];  // coalesced in b
  }
#endif
  __syncthreads();
#pragma unroll
  for (int i = 0; i < 4; ++i) {
    const int b = tb + ly + i * 8;
    const size_t o = (size_t)b * NODES + tn + lx;
    out[o] = sigmoidf(tile[lx][ly + i * 8] + in[o]);         // coalesced in n
  }
}

// ---------------- launch ----------------------------------------------------

extern "C" void kernel_launch(void* const* d_in, const int* in_sizes, int n_in,
                              void* d_out, int out_size, void* d_ws, size_t ws_size,
                              hipStream_t stream) {
  (void)in_sizes; (void)n_in; (void)out_size; (void)ws_size;

  const float* input_llr = (const float*)d_in[0];  // (1024, 8448)
  const float* w_ch      = (const float*)d_in[1];  // (8448)
  const float* w_res     = (const float*)d_in[2];  // (2, 8448)
  const int*   cidx      = (const int*)  d_in[3];  // (8448, 10)
  // d_in[4] (var_index_tensor) is unused by the reference.

  float* out = (float*)d_out;

  const size_t NB = (size_t)NODES * BATCH;         // 8.65M floats / 34.6 MB
  float* W  = (float*)d_ws;                        // weighted_llr (N,B)
  float* A0 = W  + NB;                             // rotating message buffers
  float* A1 = A0 + NB;
  float* A2 = A1 + NB;

  const dim3 blk(32, 8);
  const dim3 grd(NODES / 32, BATCH / 32);          // 264 x 32 tiles

  k_transpose_weight<<<grd, blk, 0, stream>>>(input_llr, w_ch, W);

  // prev-list schedule: it0 residual = w0*v; it>=1 residual = w0*v + w1*v_prev
  k_iter<<<NODES, 256, 0, stream>>>(W, W,  nullptr, A0, cidx, w_res, 0);
  k_iter<<<NODES, 256, 0, stream>>>(W, A0, W,       A1, cidx, w_res, 1);
  k_iter<<<NODES, 256, 0, stream>>>(W, A1, A0,      A2, cidx, w_res, 1);
  k_iter<<<NODES, 256, 0, stream>>>(W, A2, A1,      A0, cidx, w_res, 1);
  k_iter<<<NODES, 256, 0, stream>>>(W, A0, A2,      A1, cidx, w_res, 1);

  k_final<<<grd, blk, 0, stream>>>(A1, input_llr, out);
}